// GraphNeuralNetwork_88759794139603
// MI455X (gfx1250) — compile-verified
//
#include <hip/hip_runtime.h>
#include <math.h>

#define NNODES 20000
#define NEG_SLOPE 0.2f
#define K1PAD 52   // 50 rounded up to multiple of 4

typedef float v2f __attribute__((ext_vector_type(2)));
typedef float v8f __attribute__((ext_vector_type(8)));

// ---------------------------------------------------------------------------
// fp32 WMMA GEMM: C[M,Ncols] = A[M,K] * B[K,Ncols]  (row-major).
// REQUIRES: Kdim % 4 == 0 (caller zero-pads). Branch-free inner loop.
// M-edge: OOB A rows are clamped to row 0 — D row m depends only on A row m,
// so clamped garbage lands only in rows the guarded store skips.
// Block tile 64(M) x 128(N), 256 threads = 8 waves, wave = 16x64 (4 WMMA tiles).
// ---------------------------------------------------------------------------
__global__ __launch_bounds__(256) void wmma_gemm_f32(
    const float* __restrict__ A, const float* __restrict__ B,
    float* __restrict__ C, int M, int Kdim, int Ncols) {
  const int tid  = threadIdx.x;
  const int wid  = tid >> 5;
  const int lane = tid & 31;
  const int half = lane >> 4;   // 0: K pair {k0,k0+1}, 1: {k0+2,k0+3}
  const int r    = lane & 15;

  const int m0 = blockIdx.x * 64 + (wid & 3) * 16;
  const int n0 = blockIdx.y * 128 + (wid >> 2) * 64;

  v8f acc0 = {}, acc1 = {}, acc2 = {}, acc3 = {};

  const int arow = m0 + r;
  const float* __restrict__ Ap =
      A + (size_t)(arow < M ? arow : 0) * Kdim + 2 * half;   // 8B aligned (Kdim even)
  const int c0 = n0 + r;
  const float* __restrict__ Bp = B + (size_t)(2 * half) * Ncols + c0;
  const size_t ldb  = (size_t)Ncols;
  const size_t bstp = 4 * ldb;

  const int steps = Kdim >> 2;
#pragma unroll 2
  for (int it = 0; it < steps; ++it) {
    const v2f a = *(const v2f*)Ap;
    v2f b0, b1, b2, b3;
    b0.x = Bp[0];   b0.y = Bp[ldb];
    b1.x = Bp[16];  b1.y = Bp[ldb + 16];
    b2.x = Bp[32];  b2.y = Bp[ldb + 32];
    b3.x = Bp[48];  b3.y = Bp[ldb + 48];

    acc0 = __builtin_amdgcn_wmma_f32_16x16x4_f32(false, a, false, b0, (short)0, acc0, false, false);
    acc1 = __builtin_amdgcn_wmma_f32_16x16x4_f32(false, a, false, b1, (short)0, acc1, false, false);
    acc2 = __builtin_amdgcn_wmma_f32_16x16x4_f32(false, a, false, b2, (short)0, acc2, false, false);
    acc3 = __builtin_amdgcn_wmma_f32_16x16x4_f32(false, a, false, b3, (short)0, acc3, false, false);

    Ap += 4;
    Bp += bstp;
  }

#pragma unroll
  for (int v = 0; v < 8; ++v) {
    const int row = m0 + v + 8 * half;
    if (row < M) {
      float* Crow = C + (size_t)row * Ncols + n0 + r;
      Crow[0]  = acc0[v];
      Crow[16] = acc1[v];
      Crow[32] = acc2[v];
      Crow[48] = acc3[v];
    }
  }
}

// ---------------------------------------------------------------------------
// Zero-padding copies for layer-1 GEMM operands (K: 50 -> 52).
// ---------------------------------------------------------------------------
__global__ void pad_rows(const float* __restrict__ src, float* __restrict__ dst,
                         int rows, int kin, int kout) {
  const int i = blockIdx.x * blockDim.x + threadIdx.x;
  if (i >= rows * kout) return;
  const int rr = i / kout, k = i - rr * kout;
  dst[i] = (k < kin) ? src[(size_t)rr * kin + k] : 0.f;
}

__global__ void pad_w_rows(const float* __restrict__ src, float* __restrict__ dst,
                           int kin, int kout, int ncols) {
  const int i = blockIdx.x * blockDim.x + threadIdx.x;
  if (i >= kout * ncols) return;
  const int k = i / ncols;
  dst[i] = (k < kin) ? src[i] : 0.f;   // row-major [K, Ncols]: same flat index
}

// ---------------------------------------------------------------------------
// Attention logits: one wave per (node, head). 4 heads x 128 channels.
// ---------------------------------------------------------------------------
__global__ void att_scores4(const float* __restrict__ h,
                            const float* __restrict__ att_src,
                            const float* __restrict__ att_dst,
                            float* __restrict__ asrc, float* __restrict__ adst, int Nn) {
  const int gw   = (int)((blockIdx.x * blockDim.x + threadIdx.x) >> 5);
  const int lane = threadIdx.x & 31;
  const int n    = gw >> 2;
  const int head = gw & 3;
  if (n >= Nn) return;

  const float4 hv = *(const float4*)&h[(size_t)n * 512 + head * 128 + lane * 4];
  const float4 as = *(const float4*)&att_src[head * 128 + lane * 4];
  const float4 ad = *(const float4*)&att_dst[head * 128 + lane * 4];
  float ps = hv.x * as.x + hv.y * as.y + hv.z * as.z + hv.w * as.w;
  float pd = hv.x * ad.x + hv.y * ad.y + hv.z * ad.z + hv.w * ad.w;
#pragma unroll
  for (int off = 16; off; off >>= 1) {
    ps += __shfl_xor(ps, off, 32);
    pd += __shfl_xor(pd, off, 32);
  }
  if (lane == 0) {
    asrc[n * 4 + head] = ps;
    adst[n * 4 + head] = pd;
  }
}

// ---------------------------------------------------------------------------
// GEMV + attention logits for layer 3 (out_dim = 1): one wave per node.
// ---------------------------------------------------------------------------
__global__ void gemv_att3(const float* __restrict__ x3, const float* __restrict__ W3,
                          const float* __restrict__ att_s, const float* __restrict__ att_d,
                          float* __restrict__ h3, float* __restrict__ as3,
                          float* __restrict__ ad3, int Nn) {
  const int gw   = (int)((blockIdx.x * blockDim.x + threadIdx.x) >> 5);
  const int lane = threadIdx.x & 31;
  if (gw >= Nn) return;
  const float* xr = x3 + (size_t)gw * 512;
  float p = 0.f;
#pragma unroll
  for (int i = 0; i < 16; i += 4) {
    const float4 xv = *(const float4*)&xr[lane * 16 + i];
    const float4 wv = *(const float4*)&W3[lane * 16 + i];
    p += xv.x * wv.x + xv.y * wv.y + xv.z * wv.z + xv.w * wv.w;
  }
#pragma unroll
  for (int off = 16; off; off >>= 1) p += __shfl_xor(p, off, 32);
  if (lane == 0) {
    h3[gw]  = p;
    as3[gw] = p * att_s[0];
    ad3[gw] = p * att_d[0];
  }
}

// ---------------------------------------------------------------------------
// Helpers
// ---------------------------------------------------------------------------
__device__ __forceinline__ void atomicMaxF(float* addr, float v) {
  if (v >= 0.f) atomicMax((int*)addr, __float_as_int(v));
  else          atomicMin((unsigned int*)addr, __float_as_uint(v));
}
__device__ __forceinline__ float lrelu(float x) { return x > 0.f ? x : NEG_SLOPE * x; }

__global__ void fill_f32(float* __restrict__ p, float v, int n) {
  const int i = blockIdx.x * blockDim.x + threadIdx.x;
  if (i < n) p[i] = v;
}

// ---------------------------------------------------------------------------
// Edge pass 1 (H=4): segment max of leaky_relu(a_src[src]+a_dst[dst]).
// ---------------------------------------------------------------------------
__global__ void edge_max4(const long long* __restrict__ src, const long long* __restrict__ dst,
                          const float* __restrict__ asrc, const float* __restrict__ adst,
                          float* __restrict__ m, int E, int Nn) {
  const int e = blockIdx.x * blockDim.x + threadIdx.x;
  if (e >= E + Nn) return;
  int s_, d_;
  if (e < E) { s_ = (int)src[e]; d_ = (int)dst[e]; } else { s_ = d_ = e - E; }
  const float4 av = *(const float4*)&asrc[s_ * 4];
  const float4 dv = *(const float4*)&adst[d_ * 4];
  atomicMaxF(&m[d_ * 4 + 0], lrelu(av.x + dv.x));
  atomicMaxF(&m[d_ * 4 + 1], lrelu(av.y + dv.y));
  atomicMaxF(&m[d_ * 4 + 2], lrelu(av.z + dv.z));
  atomicMaxF(&m[d_ * 4 + 3], lrelu(av.w + dv.w));
}

// ---------------------------------------------------------------------------
// Edge pass 2 (H=4): one 128-thread block per edge; thread t handles channels
// 4t..4t+3 (float4), head = t>>5. Unnormalized accumulation + denominator.
// ---------------------------------------------------------------------------
__global__ __launch_bounds__(128) void edge_agg4(
    const long long* __restrict__ src, const long long* __restrict__ dst,
    const float* __restrict__ asrc, const float* __restrict__ adst,
    const float* __restrict__ m, const float* __restrict__ hfeat,
    float* __restrict__ s, float* __restrict__ out, int E, int Nn) {
  const int e = blockIdx.x;
  const int t = threadIdx.x;
  int s_, d_;
  if (e < E) { s_ = (int)src[e]; d_ = (int)dst[e]; } else { s_ = d_ = e - E; }
  const int head = t >> 5;

  const float al = lrelu(asrc[s_ * 4 + head] + adst[d_ * 4 + head]);
  const float w  = __expf(al - m[d_ * 4 + head]);
  if ((t & 31) == 0) atomicAdd(&s[d_ * 4 + head], w);

  const float4 hv = *(const float4*)&hfeat[(size_t)s_ * 512 + t * 4];
  float* o = &out[(size_t)d_ * 512 + t * 4];
  atomicAdd(o + 0, w * hv.x);
  atomicAdd(o + 1, w * hv.y);
  atomicAdd(o + 2, w * hv.z);
  atomicAdd(o + 3, w * hv.w);
}

// Epilogue layers 1/2: normalize, +bias, ELU (in place).
__global__ void finalize4(float* __restrict__ out, const float* __restrict__ s,
                          const float* __restrict__ bias, int Nn) {
  const int idx = blockIdx.x * blockDim.x + threadIdx.x;
  if (idx >= Nn * 512) return;
  const int n = idx >> 9, j = idx & 511, head = j >> 7;
  float v = out[idx] / (s[n * 4 + head] + 1e-16f) + bias[j];
  out[idx] = v > 0.f ? v : expm1f(v);
}

// ---------------------------------------------------------------------------
// Layer-3 (H=1, C=1) scalar edge passes.
// ---------------------------------------------------------------------------
__global__ void edge_max1(const long long* __restrict__ src, const long long* __restrict__ dst,
                          const float* __restrict__ as3, const float* __restrict__ ad3,
                          float* __restrict__ m3, int E, int Nn) {
  const int e = blockIdx.x * blockDim.x + threadIdx.x;
  if (e >= E + Nn) return;
  int s_, d_;
  if (e < E) { s_ = (int)src[e]; d_ = (int)dst[e]; } else { s_ = d_ = e - E; }
  atomicMaxF(&m3[d_], lrelu(as3[s_] + ad3[d_]));
}

__global__ void edge_agg1(const long long* __restrict__ src, const long long* __restrict__ dst,
                          const float* __restrict__ as3, const float* __restrict__ ad3,
                          const float* __restrict__ m3, const float* __restrict__ h3,
                          float* __restrict__ s3, float* __restrict__ o3, int E, int Nn) {
  const int e = blockIdx.x * blockDim.x + threadIdx.x;
  if (e >= E + Nn) return;
  int s_, d_;
  if (e < E) { s_ = (int)src[e]; d_ = (int)dst[e]; } else { s_ = d_ = e - E; }
  const float w = __expf(lrelu(as3[s_] + ad3[d_]) - m3[d_]);
  atomicAdd(&s3[d_], w);
  atomicAdd(&o3[d_], w * h3[s_]);
}

__global__ void finalize1(const float* __restrict__ o3, const float* __restrict__ s3,
                          const float* __restrict__ b3, float* __restrict__ out, int Nn) {
  const int n = blockIdx.x * blockDim.x + threadIdx.x;
  if (n < Nn) out[n] = o3[n] / (s3[n] + 1e-16f) + b3[0];
}

// ---------------------------------------------------------------------------
// Launcher
// ---------------------------------------------------------------------------
extern "C" void kernel_launch(void* const* d_in, const int* in_sizes, int n_in,
                              void* d_out, int out_size, void* d_ws, size_t ws_size,
                              hipStream_t stream) {
  const float*     x        = (const float*)d_in[0];
  const long long* ei       = (const long long*)d_in[1];  // int64 [2, E]
  const float*     W1       = (const float*)d_in[2];
  const float*     att_src1 = (const float*)d_in[3];
  const float*     att_dst1 = (const float*)d_in[4];
  const float*     b1       = (const float*)d_in[5];
  const float*     W2       = (const float*)d_in[6];
  const float*     att_src2 = (const float*)d_in[7];
  const float*     att_dst2 = (const float*)d_in[8];
  const float*     b2       = (const float*)d_in[9];
  const float*     W3       = (const float*)d_in[10];
  const float*     att_src3 = (const float*)d_in[11];
  const float*     att_dst3 = (const float*)d_in[12];
  const float*     b3       = (const float*)d_in[13];

  const int N = NNODES;
  const int E = in_sizes[1] / 2;
  const int Etot = E + N;
  const long long* srcIdx = ei;
  const long long* dstIdx = ei + E;

  // workspace carve-up (floats)
  float* bufA  = (float*)d_ws;                 // [N,512]
  float* bufB  = bufA + (size_t)N * 512;       // [N,512]
  float* asrc  = bufB + (size_t)N * 512;       // [N,4]
  float* adst  = asrc + (size_t)N * 4;
  float* marr  = adst + (size_t)N * 4;
  float* sarr  = marr + (size_t)N * 4;
  float* h3    = sarr + (size_t)N * 4;         // [N]
  float* as3   = h3 + N;
  float* ad3   = as3 + N;
  float* m3    = ad3 + N;
  float* s3    = m3 + N;
  float* o3    = s3 + N;
  float* xpad  = o3 + N;                       // [N, 52]
  float* w1pad = xpad + (size_t)N * K1PAD;     // [52, 512]

  const dim3 gemmGrid((N + 63) / 64, 512 / 128);
  const int  nWaveBlocks4 = (N * 4 * 32 + 255) / 256;
  const int  nWaveBlocks1 = (N * 32 + 255) / 256;
  const int  edgeBlocks   = (Etot + 255) / 256;
  const int  nh4          = N * 4;
  const int  big          = N * 512;

  // ---------------- Layer 1 ----------------
  pad_rows<<<(N * K1PAD + 255) / 256, 256, 0, stream>>>(x, xpad, N, 50, K1PAD);
  pad_w_rows<<<(K1PAD * 512 + 255) / 256, 256, 0, stream>>>(W1, w1pad, 50, K1PAD, 512);
  wmma_gemm_f32<<<gemmGrid, 256, 0, stream>>>(xpad, w1pad, bufA, N, K1PAD, 512);
  att_scores4<<<nWaveBlocks4, 256, 0, stream>>>(bufA, att_src1, att_dst1, asrc, adst, N);
  fill_f32<<<(nh4 + 255) / 256, 256, 0, stream>>>(marr, -INFINITY, nh4);
  fill_f32<<<(nh4 + 255) / 256, 256, 0, stream>>>(sarr, 0.f, nh4);
  fill_f32<<<(big + 255) / 256, 256, 0, stream>>>(bufB, 0.f, big);
  edge_max4<<<edgeBlocks, 256, 0, stream>>>(srcIdx, dstIdx, asrc, adst, marr, E, N);
  edge_agg4<<<Etot, 128, 0, stream>>>(srcIdx, dstIdx, asrc, adst, marr, bufA, sarr, bufB, E, N);
  finalize4<<<(big + 255) / 256, 256, 0, stream>>>(bufB, sarr, b1, N);

  // ---------------- Layer 2 ----------------
  wmma_gemm_f32<<<gemmGrid, 256, 0, stream>>>(bufB, W2, bufA, N, 512, 512);
  att_scores4<<<nWaveBlocks4, 256, 0, stream>>>(bufA, att_src2, att_dst2, asrc, adst, N);
  fill_f32<<<(nh4 + 255) / 256, 256, 0, stream>>>(marr, -INFINITY, nh4);
  fill_f32<<<(nh4 + 255) / 256, 256, 0, stream>>>(sarr, 0.f, nh4);
  fill_f32<<<(big + 255) / 256, 256, 0, stream>>>(bufB, 0.f, big);
  edge_max4<<<edgeBlocks, 256, 0, stream>>>(srcIdx, dstIdx, asrc, adst, marr, E, N);
  edge_agg4<<<Etot, 128, 0, stream>>>(srcIdx, dstIdx, asrc, adst, marr, bufA, sarr, bufB, E, N);
  finalize4<<<(big + 255) / 256, 256, 0, stream>>>(bufB, sarr, b2, N);

  // ---------------- Layer 3 ----------------
  gemv_att3<<<nWaveBlocks1, 256, 0, stream>>>(bufB, W3, att_src3, att_dst3, h3, as3, ad3, N);
  fill_f32<<<(N + 255) / 256, 256, 0, stream>>>(m3, -INFINITY, N);
  fill_f32<<<(N + 255) / 256, 256, 0, stream>>>(s3, 0.f, N);
  fill_f32<<<(N + 255) / 256, 256, 0, stream>>>(o3, 0.f, N);
  edge_max1<<<edgeBlocks, 256, 0, stream>>>(srcIdx, dstIdx, as3, ad3, m3, E, N);
  edge_agg1<<<edgeBlocks, 256, 0, stream>>>(srcIdx, dstIdx, as3, ad3, m3, h3, s3, o3, E, N);
  finalize1<<<(N + 255) / 256, 256, 0, stream>>>(o3, s3, b3, (float*)d_out, N);
}